// MultiScaleSparseProjection_28802050687286
// MI455X (gfx1250) — compile-verified
//
#include <hip/hip_runtime.h>
#include <hip/hip_bf16.h>

typedef __attribute__((ext_vector_type(2))) float v2f;
typedef __attribute__((ext_vector_type(8))) float v8f;

#define MPTS   400000
#define NSCL   4
#define NCH    64
#define NOUT   64
#define NBLK   12500            // MPTS / 32
#define BN_EPS 1e-5f
#define NEG_SLOPE 0.01f

// Voxel table sizes per scale: B * (128/s)^3 for s = 2,4,8,16
#define NV0 1048576L            // 4*64^3
#define NV1 131072L             // 4*32^3
#define NV2 16384L              // 4*16^3
#define NV3 2048L               // 4*8^3
#define NVOX (NV0 + NV1 + NV2 + NV3)   // 1,198,080

__device__ __forceinline__ long vox_offset(int s) {
    const long off[4] = {0L, NV0, NV0 + NV1, NV0 + NV1 + NV2};
    return off[s];
}

// ---------------------------------------------------------------------------
// Pass 1: zero voxel sums + counts (NVOX*65 floats), float4 grid-stride
// ---------------------------------------------------------------------------
__global__ void zero_vox_kernel(float4* __restrict__ p, long n4) {
    long i = (long)blockIdx.x * blockDim.x + threadIdx.x;
    long stride = (long)gridDim.x * blockDim.x;
    float4 z = {0.f, 0.f, 0.f, 0.f};
    for (; i < n4; i += stride) p[i] = z;
}

// ---------------------------------------------------------------------------
// Pass 2: voxel scatter. One thread per (point, channel); HW f32 atomics.
// ---------------------------------------------------------------------------
__global__ void voxel_accum_kernel(const float* __restrict__ feats,
                                   const int*   __restrict__ indices,
                                   float* __restrict__ vox_sum,
                                   float* __restrict__ vox_cnt) {
    long gid = (long)blockIdx.x * blockDim.x + threadIdx.x;  // M*64 threads
    int m = (int)(gid >> 6);
    int c = (int)(gid & 63);
    if (m >= MPTS) return;
    float f  = feats[gid];
    int bt = indices[m * 4 + 0];
    int px = indices[m * 4 + 1];
    int py = indices[m * 4 + 2];
    int pz = indices[m * 4 + 3];
#pragma unroll
    for (int s = 0; s < NSCL; ++s) {
        int sh = s + 1;                 // scales 2,4,8,16 -> shift 1..4
        int Gs = 128 >> sh;
        int cx = px >> sh, cy = py >> sh, cz = pz >> sh;
        long vox = vox_offset(s) +
                   ((long)((bt * Gs + cx) * Gs + cy) * Gs + cz);
        unsafeAtomicAdd(&vox_sum[vox * NCH + c], f);
        if (c == 0) unsafeAtomicAdd(&vox_cnt[vox], 1.0f);
    }
}

// ---------------------------------------------------------------------------
// Pass 3: fused center + GEMM (f32 WMMA) + BN-partial emission.
// Block = 256 threads = 8 waves. Block owns 32 rows.
// Wave w: rows [blk*32 + (w>>2)*16, +16), scale s = w & 3.
// ---------------------------------------------------------------------------
__global__ void project_wmma_kernel(const float* __restrict__ feats,
                                    const int*   __restrict__ indices,
                                    const float* __restrict__ W_proj,
                                    const float* __restrict__ b_proj,
                                    const float* __restrict__ vox_sum,
                                    const float* __restrict__ vox_cnt,
                                    float* __restrict__ out,
                                    float* __restrict__ partials) {
    __shared__ float sOs[8][16][NCH];           // 32 KB: per-wave Os tiles

    const int w    = threadIdx.x >> 5;          // wave id 0..7 (wave32)
    const int lane = threadIdx.x & 31;
    const int s    = w & 3;                     // scale index
    const int t    = w >> 2;                    // row subtile 0/1
    const int mr   = blockIdx.x * 32 + t * 16;  // first row of this wave
    const int sh   = s + 1;
    const int Gs   = 128 >> sh;
    const long voff = vox_offset(s);

    // ---- Phase A: build centered-product tile Os (16 x 64) in LDS ----
    const int c0 = lane;        // this lane's two channels
    const int c1 = lane + 32;
    for (int r = 0; r < 16; ++r) {
        int m = mr + r;
        int bt = indices[m * 4 + 0];
        int cx = indices[m * 4 + 1] >> sh;
        int cy = indices[m * 4 + 2] >> sh;
        int cz = indices[m * 4 + 3] >> sh;
        long vox = voff + ((long)((bt * Gs + cx) * Gs + cy) * Gs + cz);
        float cnt  = vox_cnt[vox];
        float rinv = 1.0f / fmaxf(cnt, 1.0f);
        const float* vs = vox_sum + vox * NCH;
        const float* fr = feats + (long)m * NCH;
        float f0 = fr[c0], f1 = fr[c1];
        sOs[w][r][c0] = (f0 - vs[c0] * rinv) * f0;
        sOs[w][r][c1] = (f1 - vs[c1] * rinv) * f1;
    }
    __syncthreads();

    // ---- Phase B: h(16x64) = Os(16x64) @ W[s](64x64) + bias, via WMMA ----
    const int half = lane >> 4;                 // 0: lanes 0-15, 1: 16-31
    const int l15  = lane & 15;
    const float* Wst = W_proj + (long)s * NCH * NOUT;   // row-major [k][n]

    for (int nt = 0; nt < 4; ++nt) {
        const int n = nt * 16 + l15;            // output column (C/D: N=lane&15)
        const float bias = b_proj[s * NOUT + n];
        v8f acc;
#pragma unroll
        for (int j = 0; j < 8; ++j) acc[j] = bias;

#pragma unroll
        for (int k0 = 0; k0 < NCH; k0 += 4) {
            const int ka = k0 + 2 * half;       // A/B VGPR0 K, VGPR1 K = ka+1
            v2f a, b;
            a.x = sOs[w][l15][ka];              // A: M = lane&15
            a.y = sOs[w][l15][ka + 1];
            b.x = Wst[ka * NOUT + n];           // B: row K, col N
            b.y = Wst[(ka + 1) * NOUT + n];
            acc = __builtin_amdgcn_wmma_f32_16x16x4_f32(
                false, a, false, b, (short)0, acc, false, false);
        }

        // Store h tile and accumulate BN partials (per-column sum / sumsq)
        float cs = 0.f, cq = 0.f;
#pragma unroll
        for (int j = 0; j < 8; ++j) {
            int m = mr + j + 8 * half;          // C/D: M = j (+8 for hi half)
            out[((long)m * NSCL + s) * NOUT + n] = acc[j];
            cs += acc[j];
            cq += acc[j] * acc[j];
        }
        // Combine the two lane-halves (same column n) across the wave32
        cs += __shfl_xor(cs, 16, 32);
        cq += __shfl_xor(cq, 16, 32);
        if (half == 0) {
            long row = ((long)blockIdx.x * 8 + w) * 128;
            partials[row + n]      = cs;
            partials[row + 64 + n] = cq;
        }
    }
}

// ---------------------------------------------------------------------------
// Pass 4: deterministic reduction of partials -> bnstats[s][stat][o]
// 512 outputs; each thread serially sums its series (no float atomics).
// ---------------------------------------------------------------------------
__global__ void bn_reduce_kernel(const float* __restrict__ partials,
                                 float* __restrict__ bnstats) {
    int tid = blockIdx.x * blockDim.x + threadIdx.x;
    if (tid >= NSCL * 128) return;
    int s   = tid >> 7;                 // scale
    int idx = tid & 127;                // stat*64 + o (matches partial row)
    float acc = 0.f;
    for (int blk = 0; blk < NBLK; ++blk) {
#pragma unroll
        for (int t = 0; t < 2; ++t) {
            acc += partials[((long)blk * 8 + t * 4 + s) * 128 + idx];
        }
    }
    bnstats[tid] = acc;                 // bnstats[s*128 + stat*64 + o]
}

// ---------------------------------------------------------------------------
// Pass 5: BN + LeakyReLU, in place on d_out, float4-vectorized.
// ---------------------------------------------------------------------------
__global__ void bn_apply_kernel(float* __restrict__ out,
                                const float* __restrict__ bnstats,
                                const float* __restrict__ gamma,
                                const float* __restrict__ beta) {
    long e = ((long)blockIdx.x * blockDim.x + threadIdx.x) * 4;
    const long total = (long)MPTS * NSCL * NOUT;
    if (e >= total) return;
    float4 h = *(const float4*)(out + e);
    int s  = (int)((e >> 6) & 3);
    int o0 = (int)(e & 63);
    const float invM = 1.0f / (float)MPTS;
    float r[4] = {h.x, h.y, h.z, h.w};
#pragma unroll
    for (int i = 0; i < 4; ++i) {
        int o = o0 + i;
        float mu  = bnstats[s * 128 + o] * invM;
        float msq = bnstats[s * 128 + 64 + o] * invM;
        float var = msq - mu * mu;
        float sc  = __frsqrt_rn(var + BN_EPS) * gamma[s * NOUT + o];
        float v   = (r[i] - mu) * sc + beta[s * NOUT + o];
        r[i] = v > 0.f ? v : NEG_SLOPE * v;
    }
    float4 res = {r[0], r[1], r[2], r[3]};
    *(float4*)(out + e) = res;
}

// ---------------------------------------------------------------------------
extern "C" void kernel_launch(void* const* d_in, const int* in_sizes, int n_in,
                              void* d_out, int out_size, void* d_ws, size_t ws_size,
                              hipStream_t stream) {
    const float* feats   = (const float*)d_in[0];   // (M, 64) f32
    const int*   indices = (const int*)  d_in[1];   // (M, 4)  i32
    const float* W_proj  = (const float*)d_in[2];   // (4, 64, 64) f32
    const float* b_proj  = (const float*)d_in[3];   // (4, 64) f32
    const float* gamma   = (const float*)d_in[4];   // (4, 64) f32
    const float* beta    = (const float*)d_in[5];   // (4, 64) f32
    float* out = (float*)d_out;                     // (M, 4, 64) f32
    float* ws  = (float*)d_ws;

    float* vox_sum  = ws;                                   // NVOX * 64
    float* vox_cnt  = ws + NVOX * 64;                       // NVOX
    float* partials = ws + NVOX * 65;                       // NBLK*8*128
    float* bnstats  = partials + (long)NBLK * 8 * 128;      // 512

    // Pass 1: zero voxel tables (NVOX*65 floats, divisible by 4)
    zero_vox_kernel<<<4096, 256, 0, stream>>>((float4*)ws, NVOX * 65 / 4);

    // Pass 2: voxel scatter (M*64 threads)
    voxel_accum_kernel<<<(MPTS * 64) / 256, 256, 0, stream>>>(
        feats, indices, vox_sum, vox_cnt);

    // Pass 3: fused center + WMMA GEMM + BN partials (32 rows per block)
    project_wmma_kernel<<<NBLK, 256, 0, stream>>>(
        feats, indices, W_proj, b_proj, vox_sum, vox_cnt, out, partials);

    // Pass 4: deterministic BN statistic reduction (512 outputs)
    bn_reduce_kernel<<<2, 256, 0, stream>>>(partials, bnstats);

    // Pass 5: BN + LeakyReLU in place (float4 per thread)
    bn_apply_kernel<<<(MPTS * NSCL * NOUT / 4) / 256, 256, 0, stream>>>(
        out, bnstats, gamma, beta);
}